// EfficientGNN_15298673509049
// MI455X (gfx1250) — compile-verified
//
#include <hip/hip_runtime.h>

// CDNA5 / gfx1250, wave32.
typedef __attribute__((ext_vector_type(2))) float v2f;
typedef __attribute__((ext_vector_type(8))) float v8f;

#define HIDDEN 128  // reference HIDDEN

// ---------------------------------------------------------------------------
// K1: init workspace. deg starts at 1.0 (self-loop), s1/w/v zeroed.
// ---------------------------------------------------------------------------
__global__ void gcn_init(float* __restrict__ deg, float* __restrict__ s1,
                         float* __restrict__ w, float* __restrict__ v, int n) {
  int i = blockIdx.x * blockDim.x + threadIdx.x;
  if (i < n) {
    deg[i] = 1.0f;   // self-loop contributes 1 to destination degree
    s1[i]  = 0.0f;
    w[i]   = 0.0f;
  }
  if (i < HIDDEN) v[i] = 0.0f;
}

// ---------------------------------------------------------------------------
// K2: degree over destinations: deg[col[e]] += 1  (float adds of 1.0 exact)
// ---------------------------------------------------------------------------
__global__ void gcn_degree(const int* __restrict__ col, float* __restrict__ deg, int E) {
  int e = blockIdx.x * blockDim.x + threadIdx.x;
  if (e < E) atomicAdd(&deg[col[e]], 1.0f);
}

// ---------------------------------------------------------------------------
// K3: dinv = rsqrt(deg) (in place) and fold self-loop terms:
//     s1[i] += dinv[i]^2 * x[i]   (self edge norm = dinv*dinv)
//     w[i]  += dinv[i]^2
// ---------------------------------------------------------------------------
__global__ void gcn_selfloop(const float* __restrict__ x, float* __restrict__ dinv,
                             float* __restrict__ s1, float* __restrict__ w, int n) {
  int i = blockIdx.x * blockDim.x + threadIdx.x;
  if (i < n) {
    float di = rsqrtf(dinv[i]);
    dinv[i] = di;
    float sq = di * di;
    s1[i] = sq * x[i];
    w[i]  = sq;
  }
}

// ---------------------------------------------------------------------------
// K4: edge pass. norm = dinv[r]*dinv[c];
//     s1[c] += norm * x[r]   (layer-1 aggregation, collapsed to scalar)
//     w[r]  += norm          (layer-2+mean collapsed source weight)
// ---------------------------------------------------------------------------
__global__ void gcn_edges(const int* __restrict__ row, const int* __restrict__ col,
                          const float* __restrict__ x, const float* __restrict__ dinv,
                          float* __restrict__ s1, float* __restrict__ w, int E) {
  int e = blockIdx.x * blockDim.x + threadIdx.x;
  if (e < E) {
    int r = row[e];
    int c = col[e];
    float nrm = dinv[r] * dinv[c];
    atomicAdd(&s1[c], nrm * x[r]);
    atomicAdd(&w[r], nrm);
  }
}

// ---------------------------------------------------------------------------
// K5: v[f] = sum_r w[r] * relu(s1[r]*W1[f] + b1[f]),  f = 0..127.
// 128 threads/block, thread t owns feature t; s1/w loads are wave-uniform
// (scalar-load friendly). Partial sums combined with float atomics into v.
// ---------------------------------------------------------------------------
__global__ void gcn_reduce(const float* __restrict__ s1, const float* __restrict__ w,
                           const float* __restrict__ W1, const float* __restrict__ b1,
                           float* __restrict__ v, int n) {
  int f = threadIdx.x;            // 0..127
  float w1f = W1[f];
  float b1f = b1[f];
  float acc = 0.0f;
  int per   = (n + gridDim.x - 1) / gridDim.x;
  int start = blockIdx.x * per;
  int end   = start + per;
  if (end > n) end = n;
  for (int r = start; r < end; ++r) {
    float sr = s1[r];
    float wr = w[r];
    acc += wr * fmaxf(fmaf(sr, w1f, b1f), 0.0f);
  }
  atomicAdd(&v[f], acc);
}

// ---------------------------------------------------------------------------
// K6: out[o] = (1/N) * sum_f v[f]*W2[f,o] + b2[o]  via V_WMMA_F32_16X16X4_F32.
// One wave per 16-column output tile (400/16 = 25 blocks).
// A (16x4 f32, 2 VGPRs): lanes 0-15 hold rows M, K={0,1}; lanes 16-31 K={2,3}.
//   We place v[k0..k0+3] in row M=0, rows 1..15 zero.
// B (4x16 f32, 2 VGPRs): VGPR0 = rows K=0 (lanes 0-15) / K=2 (lanes 16-31),
//   VGPR1 = rows K=1 / K=3 — i.e. lane group selects krow = (lane>>4)*2.
// D (16x16 f32, 8 VGPRs): row M=0 lives in c[0], lanes 0-15.
// ---------------------------------------------------------------------------
__global__ void gcn_final_wmma(const float* __restrict__ v, const float* __restrict__ W2,
                               const float* __restrict__ b2, float* __restrict__ out,
                               int n_nodes, int out_dim) {
  int lane = threadIdx.x;          // 0..31, wave32
  int n0   = blockIdx.x * 16;      // output column tile base
  int krow = (lane >> 4) * 2;      // 0 for lanes 0-15, 2 for lanes 16-31
  int ncol = lane & 15;

  v8f c = {0.f, 0.f, 0.f, 0.f, 0.f, 0.f, 0.f, 0.f};

  for (int k0 = 0; k0 < HIDDEN; k0 += 4) {
    v2f a;
    if (ncol == 0) {               // only row M=0 of A is non-zero
      a.x = v[k0 + krow];
      a.y = v[k0 + krow + 1];
    } else {
      a.x = 0.0f;
      a.y = 0.0f;
    }
    v2f b;
    b.x = W2[(k0 + krow)     * out_dim + n0 + ncol];
    b.y = W2[(k0 + krow + 1) * out_dim + n0 + ncol];

    // 8-arg pattern: (neg_a, A, neg_b, B, c_mod, C, reuse_a, reuse_b)
    c = __builtin_amdgcn_wmma_f32_16x16x4_f32(false, a, false, b, (short)0, c,
                                              false, false);
  }

  if (lane < 16) {
    float invN = 1.0f / (float)n_nodes;
    out[n0 + lane] = c[0] * invN + b2[n0 + lane];
  }
}

// ---------------------------------------------------------------------------
// Launcher
// ---------------------------------------------------------------------------
extern "C" void kernel_launch(void* const* d_in, const int* in_sizes, int n_in,
                              void* d_out, int out_size, void* d_ws, size_t ws_size,
                              hipStream_t stream) {
  const float* x  = (const float*)d_in[0];   // [N,1]
  const int*   ei = (const int*)  d_in[1];   // [2,E] row-major
  const float* W1 = (const float*)d_in[2];   // [1,128]
  const float* b1 = (const float*)d_in[3];   // [128]
  const float* W2 = (const float*)d_in[4];   // [128,400] row-major
  const float* b2 = (const float*)d_in[5];   // [400]
  float* out = (float*)d_out;                // [400]

  const int N   = in_sizes[0];
  const int E   = in_sizes[1] / 2;
  const int OUT = in_sizes[5];

  const int* row = ei;        // sources
  const int* col = ei + E;    // destinations

  // Workspace: deg/dinv (N) | s1 (N) | w (N) | v (HIDDEN)   ~1.2 MB
  float* ws       = (float*)d_ws;
  float* deg_dinv = ws;
  float* s1       = ws + (size_t)N;
  float* w        = ws + (size_t)2 * N;
  float* v        = ws + (size_t)3 * N;

  const int TB = 256;
  gcn_init    <<<(N + TB - 1) / TB, TB, 0, stream>>>(deg_dinv, s1, w, v, N);
  gcn_degree  <<<(E + TB - 1) / TB, TB, 0, stream>>>(col, deg_dinv, E);
  gcn_selfloop<<<(N + TB - 1) / TB, TB, 0, stream>>>(x, deg_dinv, s1, w, N);
  gcn_edges   <<<(E + TB - 1) / TB, TB, 0, stream>>>(row, col, x, deg_dinv, s1, w, E);
  gcn_reduce  <<<256, HIDDEN, 0, stream>>>(s1, w, W1, b1, v, N);
  gcn_final_wmma<<<OUT / 16, 32, 0, stream>>>(v, W2, b2, out, N, OUT);
}